// TSPTourEncoder_9397388444095
// MI455X (gfx1250) — compile-verified
//
#include <hip/hip_runtime.h>
#include <hip/hip_bf16.h>

typedef __attribute__((ext_vector_type(2))) float v2f;
typedef __attribute__((ext_vector_type(8))) float v8f;

#define EMPTY_KEY 0xFFFFFFFFFFFFFFFFull
#define VAL_EMPTY 0x7FFFFFFF

__device__ __forceinline__ unsigned int hash_slot(unsigned long long key, int tabBits) {
    return (unsigned int)((key * 0x9E3779B97F4A7C15ull) >> (64 - tabBits));
}

// ---------------------------------------------------------------- hash init
__global__ void ht_init_kernel(unsigned long long* __restrict__ keys,
                               int* __restrict__ vals, int slots) {
    int i = blockIdx.x * blockDim.x + threadIdx.x;
    if (i < slots) {
        keys[i] = EMPTY_KEY;
        vals[i] = VAL_EMPTY;
    }
}

// ---------------------------------------------------------------- hash build
// key = src*M + dst  ->  edge id.  Duplicate keys keep the SMALLEST edge id,
// matching the reference's stable argsort + searchsorted('left') behavior.
__global__ void ht_build_kernel(const long long* __restrict__ edge_index, int E,
                                long long M,
                                unsigned long long* __restrict__ keys,
                                int* __restrict__ vals, int tabBits) {
    int e = blockIdx.x * blockDim.x + threadIdx.x;
    if (e >= E) return;
    unsigned long long s = (unsigned long long)edge_index[e];
    unsigned long long d = (unsigned long long)edge_index[E + e];
    unsigned long long key = s * (unsigned long long)M + d;
    unsigned int mask = (1u << tabBits) - 1u;
    unsigned int h = hash_slot(key, tabBits);
    for (;;) {
        unsigned long long prev = atomicCAS(&keys[h], EMPTY_KEY, key);
        if (prev == EMPTY_KEY || prev == key) {
            atomicMin(&vals[h], e);
            return;
        }
        h = (h + 1u) & mask;
    }
}

__device__ __forceinline__ int ht_lookup(unsigned long long key,
                                         const unsigned long long* __restrict__ keys,
                                         const int* __restrict__ vals, int tabBits) {
    unsigned int mask = (1u << tabBits) - 1u;
    unsigned int h = hash_slot(key, tabBits);
    for (;;) {
        unsigned long long k = keys[h];
        if (k == key) return vals[h];
        if (k == EMPTY_KEY) return -1;
        h = (h + 1u) & mask;
    }
}

// ---------------------------------------------------------------- main kernel
// One block per (s,b).  Phase 1: hash lookups -> LDS arrays:
//   s_w[i]   = found ? 1/N : 0                    (goes into A-matrix row 0)
//   s_off[i] = 32-bit BYTE offset of edge_emb row (0 if missing; weight=0)
// Phase 2: masked mean as OUT(1xEM) = W(1xN) x EMB(NxEM) with
// V_WMMA_F32_16X16X4_F32.  Branch-free A build (uniform ds_load_b64 +
// v_cndmask), saddr+u32-voffset global loads, dual accumulators.
__global__ void tour_encode_kernel(const int* __restrict__ y,
                                   const long long* __restrict__ node_offset,
                                   const float* __restrict__ edge_emb,
                                   float* __restrict__ out,
                                   int N, int Npad, int EM, long long M,
                                   const unsigned long long* __restrict__ hkeys,
                                   const int* __restrict__ hvals, int tabBits) {
    extern __shared__ char smem[];
    float*        s_w   = (float*)smem;                 // Npad weights
    unsigned int* s_off = (unsigned int*)(smem + Npad * 4);  // Npad byte offsets

    const int sb = blockIdx.x;

    // ---- phase 1: per-tour-edge lookup -------------------------------------
    for (int i = threadIdx.x; i < Npad; i += blockDim.x) {
        float        wgt = 0.0f;
        unsigned int off = 0u;
        if (i < N) {
            long long base = (long long)sb * N;
            long long gs = node_offset[base + y[base + i]];
            int ip1 = (i + 1 == N) ? 0 : (i + 1);
            long long gd = node_offset[base + y[base + ip1]];
            unsigned long long kf =
                (unsigned long long)gs * (unsigned long long)M + (unsigned long long)gd;
            int v = ht_lookup(kf, hkeys, hvals, tabBits);
            if (v < 0) {
                unsigned long long kr =
                    (unsigned long long)gd * (unsigned long long)M + (unsigned long long)gs;
                v = ht_lookup(kr, hkeys, hvals, tabBits);
            }
            if (v >= 0 && v != VAL_EMPTY) {
                wgt = 1.0f / (float)N;
                off = (unsigned int)(v * EM) * 4u;   // byte offset of row
            }
        }
        s_w[i]   = wgt;
        s_off[i] = off;
    }
    __syncthreads();

    // ---- phase 2: WMMA masked-mean reduction --------------------------------
    const int      lane   = threadIdx.x & 31;
    const int      wave   = threadIdx.x >> 5;
    const int      nwaves = blockDim.x >> 5;
    const int      lane15 = lane & 15;
    // A layout (16x4 f32): lanes 0-15 hold K0(V0)/K1(V1), lanes 16-31 K2/K3.
    const int      koff   = (lane & 16) ? 2 : 0;
    const bool     act    = (lane15 == 0);  // only row M=0 is non-zero
    const char*    ebase  = (const char*)edge_emb;

    for (int cb = wave * 16; cb < EM; cb += nwaves * 16) {
        const unsigned int col4 = (unsigned int)(cb + lane15) * 4u;
        v8f acc0 = {};
        v8f acc1 = {};
        for (int k0 = 0; k0 < Npad; k0 += 8) {
            // ---- step A: K rows [k0, k0+4) -> acc0
            float2 wp0 = *(const float2*)(s_w + k0 + koff);
            uint2  op0 = *(const uint2*)(s_off + k0 + koff);
            // ---- step B: K rows [k0+4, k0+8) -> acc1
            float2 wp1 = *(const float2*)(s_w + k0 + 4 + koff);
            uint2  op1 = *(const uint2*)(s_off + k0 + 4 + koff);

            v2f b0;
            b0[0] = __builtin_nontemporal_load((const float*)(ebase + (op0.x + col4)));
            b0[1] = __builtin_nontemporal_load((const float*)(ebase + (op0.y + col4)));
            v2f b1;
            b1[0] = __builtin_nontemporal_load((const float*)(ebase + (op1.x + col4)));
            b1[1] = __builtin_nontemporal_load((const float*)(ebase + (op1.y + col4)));

            v2f a0;
            a0[0] = act ? wp0.x : 0.0f;
            a0[1] = act ? wp0.y : 0.0f;
            v2f a1;
            a1[0] = act ? wp1.x : 0.0f;
            a1[1] = act ? wp1.y : 0.0f;

            acc0 = __builtin_amdgcn_wmma_f32_16x16x4_f32(
                false, a0, false, b0, (short)0, acc0, false, false);
            acc1 = __builtin_amdgcn_wmma_f32_16x16x4_f32(
                false, a1, false, b1, (short)0, acc1, false, false);
        }
        // D row M=0 lives in element 0, lanes 0-15.
        if (lane < 16) {
            out[(long long)sb * EM + cb + lane] = acc0[0] + acc1[0];
        }
    }
}

// ---------------------------------------------------------------- launcher
extern "C" void kernel_launch(void* const* d_in, const int* in_sizes, int n_in,
                              void* d_out, int out_size, void* d_ws, size_t ws_size,
                              hipStream_t stream) {
    const int*       y           = (const int*)d_in[0];        // int32 (S,B,N)
    const long long* node_offset = (const long long*)d_in[1];  // int64 (S,B,N)
    const long long* edge_index  = (const long long*)d_in[2];  // int64 (2,E)
    const float*     edge_emb    = (const float*)d_in[3];      // f32  (E,EM)
    float*           out         = (float*)d_out;              // f32  (S,B,EM)

    const long long M  = in_sizes[1];          // total nodes (= S*B*N)
    const int       E  = in_sizes[2] / 2;      // number of edges
    const int       EM = in_sizes[3] / E;      // embedding dim (128)
    const int       SB = out_size / EM;        // S*B rows (2048)
    const int       N  = (int)(M / SB);        // tour length (100)
    const int       Npad = (N + 7) & ~7;       // dual-acc step of 8

    // Hash table in workspace: 8B key + 4B val per slot. Prefer 2^21 slots
    // (load factor ~0.2), shrink if ws is small; 2^19 still holds E=409600.
    int tabBits = 21;
    while (tabBits > 19 && ((size_t)12u << tabBits) > ws_size) tabBits--;
    const int slots = 1 << tabBits;
    unsigned long long* hkeys = (unsigned long long*)d_ws;
    int* hvals = (int*)(hkeys + slots);

    ht_init_kernel<<<(slots + 255) / 256, 256, 0, stream>>>(hkeys, hvals, slots);
    ht_build_kernel<<<(E + 255) / 256, 256, 0, stream>>>(edge_index, E, M,
                                                         hkeys, hvals, tabBits);

    const size_t lds_bytes = (size_t)Npad * 8;  // weights + byte offsets
    tour_encode_kernel<<<SB, 256, lds_bytes, stream>>>(
        y, node_offset, edge_emb, out, N, Npad, EM, M, hkeys, hvals, tabBits);
}